// RNNMemoryModel_47485158425099
// MI455X (gfx1250) — compile-verified
//
#include <hip/hip_runtime.h>
#include <hip/hip_bf16.h>
#include <stdint.h>

// ---------------- problem constants (from reference) ----------------
#define NN       1024        // neurons
#define IDIM     24          // input dim
#define KAUG     (NN + IDIM) // augmented K = 1048
#define BATCH    64
#define SEQ      500
#define NWG      16          // persistent workgroups, 64 neurons each
#define TILE_N   64
#define CH       64          // K-chunk staged in LDS (double buffered)
#define NCH      16          // full chunks (16*64 = 1024), + 24-wide tail
#define SASTR    68          // padded chunk row stride (dwords): 8*68 % 64 = 32
#define SW_ELEMS (KAUG * TILE_N)        // 67072 floats = 268288 B
#define SAB_ELEMS (BATCH * SASTR)       // 4352 floats  =  17408 B
#define LDS_BYTES ((SW_ELEMS + 2 * SAB_ELEMS) * 4)  // 303104 B <= 320KB

typedef __attribute__((ext_vector_type(2))) float v2f;
typedef __attribute__((ext_vector_type(8))) float v8f;

// phi(x) = 30*(1+tanh(0.14x-4.2)) = 60 * t/(t+1),  t = exp(0.28x - 8.4)
__device__ __forceinline__ float phi_f(float x) {
    float t = __expf(__builtin_fmaf(0.28f, x, -8.4f));
    return 60.0f * t / (t + 1.0f);
}

// ---- CDNA5 async global->LDS (ASYNCcnt) ----
__device__ __forceinline__ void async_b128(uint32_t lds_byte_off, const float* g) {
    asm volatile("global_load_async_to_lds_b128 %0, %1, off"
                 :: "v"(lds_byte_off), "v"(g) : "memory");
}
__device__ __forceinline__ void async_b64(uint32_t lds_byte_off, const float* g) {
    asm volatile("global_load_async_to_lds_b64 %0, %1, off"
                 :: "v"(lds_byte_off), "v"(g) : "memory");
}
__device__ __forceinline__ void wait_async0() {
    asm volatile("s_wait_asynccnt 0x0" ::: "memory");
}

// ---- pipelined WMMA over one staged chunk (NG groups of K=4) ----
template <int NG>
__device__ __forceinline__ void mm_chunk(const float* __restrict__ aPtr,
                                         const float* __restrict__ bPtr,
                                         v8f& acc0, v8f& acc1) {
    v2f a  = *(const v2f*)(aPtr);
    v2f b0 = *(const v2f*)(bPtr);
    v2f b1 = *(const v2f*)(bPtr + 64);
#pragma unroll
    for (int g = 0; g < NG; ++g) {
        v2f an = a, b0n = b0, b1n = b1;
        if (g + 1 < NG) {                       // prefetch next group's frags
            an  = *(const v2f*)(aPtr + (g + 1) * 4);
            b0n = *(const v2f*)(bPtr + (g + 1) * 256);
            b1n = *(const v2f*)(bPtr + (g + 1) * 256 + 64);
        }
        acc0 = __builtin_amdgcn_wmma_f32_16x16x4_f32(
            false, a, false, b0, (short)0, acc0, false, false);
        acc1 = __builtin_amdgcn_wmma_f32_16x16x4_f32(
            false, a, false, b1, (short)0, acc1, false, false);
        a = an; b0 = b0n; b1 = b1n;
    }
}

// ---------------- init: state buffer 0 = [r0 ; u_0], reset barrier ----------------
__global__ void rnn_init_kernel(const float* __restrict__ r0,
                                const float* __restrict__ u,
                                float* __restrict__ ws) {
    float* buf0 = ws;
    int b = blockIdx.x;
    for (int j = threadIdx.x; j < NN; j += blockDim.x)
        buf0[b * KAUG + j] = r0[b * NN + j];
    for (int i = threadIdx.x; i < IDIM; i += blockDim.x)
        buf0[b * KAUG + NN + i] = u[(size_t)b * (SEQ * IDIM) + i];  // t = 0
    if (b == 0 && threadIdx.x == 0)
        *(unsigned*)(ws + 2 * (size_t)KAUG * BATCH) = 0u;
}

// ---------------- persistent RNN kernel ----------------
__global__ void __launch_bounds__(256, 1)
rnn_persistent_kernel(const float* __restrict__ u,
                      const float* __restrict__ W,
                      const float* __restrict__ Bm,
                      const float* __restrict__ tau,
                      const float* __restrict__ dsign,
                      float* __restrict__ out,
                      float* __restrict__ ws) {
    extern __shared__ float smem[];
    float* sW  = smem;                     // packed W_eff tile, B-fragment layout
    float* sA0 = smem + SW_ELEMS;          // chunk double buffer
    float* sA1 = sA0 + SAB_ELEMS;
    const uint32_t sA0_base = (uint32_t)(uintptr_t)sA0;  // LDS byte offset
    const uint32_t sA1_base = (uint32_t)(uintptr_t)sA1;

    float* buf0 = ws;
    float* buf1 = ws + (size_t)KAUG * BATCH;
    unsigned* cnt = (unsigned*)(ws + 2 * (size_t)KAUG * BATCH);

    const int tid  = threadIdx.x;
    const int wg   = blockIdx.x;
    const int n0   = wg * TILE_N;
    const int wv   = tid >> 5;
    const int lane = tid & 31;
    const int ln   = lane & 15;
    const int hi2  = (lane >> 4) * 2;          // 0 | 2
    const int mi   = wv & 3;                   // batch-row subtile
    const int nbase = (wv >> 2) * 32;          // wave's 32 columns

    // ---- preload + pack W_eff tile (rows 1024..1047 = B^T) ----
    for (int idx = tid; idx < SW_ELEMS; idx += 256) {
        int k = idx >> 6;
        int n = idx & 63;
        int j = n0 + n;
        float v;
        if (k < NN) v = fmaxf(W[(size_t)k * NN + j], 0.0f) * dsign[k];
        else        v = Bm[(size_t)j * IDIM + (k - NN)];
        sW[(((k >> 2) * TILE_N + n) << 2) + (k & 3)] = v;
    }

    const float alpha0 = 0.1f / tau[n0 + nbase + ln];
    const float alpha1 = 0.1f / tau[n0 + nbase + 16 + ln];

    // per-wave fragment offsets (dwords)
    const int aOff = (mi * 16 + ln) * SASTR + hi2;
    const int bOff = ((nbase + ln) << 2) + hi2;

    __syncthreads();

    for (int t = 0; t < SEQ; ++t) {
        const float* rbuf = (t & 1) ? buf1 : buf0;
        float*       wbuf = (t & 1) ? buf0 : buf1;

        v8f acc0 = {};
        v8f acc1 = {};

        // ---- prologue: async-stage chunk 0 into sA0 ----
        for (int i = tid; i < BATCH * (CH / 4); i += 256) {
            int row = i >> 4, c4 = i & 15;
            async_b128(sA0_base + (uint32_t)(row * SASTR + c4 * 4) * 4u,
                       rbuf + (size_t)row * KAUG + c4 * 4);
        }
        wait_async0();
        __syncthreads();

        // ---- 16 full chunks + 24-wide tail, staged async one ahead ----
        for (int c = 0; c <= NCH; ++c) {
            // stage chunk c+1 into the other buffer (no wait yet)
            if (c + 1 < NCH) {
                const uint32_t dstb = ((c + 1) & 1) ? sA1_base : sA0_base;
                const int k0 = (c + 1) * CH;
                for (int i = tid; i < BATCH * (CH / 4); i += 256) {
                    int row = i >> 4, c4 = i & 15;
                    async_b128(dstb + (uint32_t)(row * SASTR + c4 * 4) * 4u,
                               rbuf + (size_t)row * KAUG + k0 + c4 * 4);
                }
            } else if (c + 1 == NCH) {         // tail chunk: the u_t part, 24 wide
                const uint32_t dstb = ((c + 1) & 1) ? sA1_base : sA0_base;
                for (int i = tid; i < BATCH * (IDIM / 2); i += 256) {
                    int row = i / 12, c2 = i - row * 12;
                    async_b64(dstb + (uint32_t)(row * SASTR + c2 * 2) * 4u,
                              rbuf + (size_t)row * KAUG + NN + c2 * 2);
                }
            }

            // compute current chunk
            const float* sAc = (c & 1) ? sA1 : sA0;
            if (c < NCH) {
                mm_chunk<CH / 4>(sAc + aOff, sW + ((c * NCH * 64) << 2) + bOff,
                                 acc0, acc1);
            } else {
                mm_chunk<IDIM / 4>(sAc + aOff, sW + ((256 * 64) << 2) + bOff,
                                   acc0, acc1);
            }

            wait_async0();
            __syncthreads();
        }

        // ---- async-stage r_old tile [64 x 64] (our columns) into sA0 ----
        for (int i = tid; i < BATCH * (TILE_N / 4); i += 256) {
            int row = i >> 4, c4 = i & 15;
            async_b128(sA0_base + (uint32_t)(row * SASTR + c4 * 4) * 4u,
                       rbuf + (size_t)row * KAUG + n0 + c4 * 4);
        }
        wait_async0();
        __syncthreads();

        // ---- activation + leaky update, in-place in sA0 ----
        // D layout: VGPR g -> (M = g + 8*hi, N = ln) within a 16x16 subtile
#pragma unroll
        for (int g = 0; g < 8; ++g) {
            int brow = mi * 16 + g + (hi2 << 2);     // + 8*hi
            {
                float* p = sA0 + brow * SASTR + nbase + ln;
                float rold = *p;
                *p = __builtin_fmaf(alpha0, phi_f(acc0[g]) - rold, rold);
            }
            {
                float* p = sA0 + brow * SASTR + nbase + 16 + ln;
                float rold = *p;
                *p = __builtin_fmaf(alpha1, phi_f(acc1[g]) - rold, rold);
            }
        }
        __syncthreads();

        // ---- coalesced writeback: next state buffer + r_out (+ r_final) ----
        for (int i = tid; i < BATCH * TILE_N; i += 256) {
            int row = i >> 6, col = i & 63;
            float v = sA0[row * SASTR + col];
            wbuf[(size_t)row * KAUG + n0 + col] = v;
            out[(size_t)row * (SEQ * NN) + (size_t)t * NN + n0 + col] = v;
            if (t == SEQ - 1)
                out[(size_t)SEQ * BATCH * NN + (size_t)row * NN + n0 + col] = v;
        }

        // ---- WG0 stages u_{t+1} into next buffer's augmented tail ----
        if (wg == 0 && (t + 1) < SEQ) {
            for (int i = tid; i < BATCH * IDIM; i += 256) {
                int row = i / IDIM, col = i - row * IDIM;
                wbuf[(size_t)row * KAUG + NN + col] =
                    u[(size_t)row * (SEQ * IDIM) + (size_t)(t + 1) * IDIM + col];
            }
        }

        // ---- device-wide barrier (release counter, s_sleep spin) ----
        __threadfence();
        __syncthreads();
        if (tid == 0) {
            __hip_atomic_fetch_add(cnt, 1u, __ATOMIC_RELEASE, __HIP_MEMORY_SCOPE_AGENT);
            const unsigned target = (unsigned)(NWG * (t + 1));
            while (__hip_atomic_load(cnt, __ATOMIC_ACQUIRE, __HIP_MEMORY_SCOPE_AGENT) < target) {
                __builtin_amdgcn_s_sleep(1);
            }
        }
        __syncthreads();
    }
}

// ---------------- host launch ----------------
extern "C" void kernel_launch(void* const* d_in, const int* in_sizes, int n_in,
                              void* d_out, int out_size, void* d_ws, size_t ws_size,
                              hipStream_t stream) {
    (void)in_sizes; (void)n_in; (void)out_size; (void)ws_size;
    const float* u     = (const float*)d_in[0];  // [64,500,24]
    const float* r0    = (const float*)d_in[1];  // [1,64,1024]
    const float* W     = (const float*)d_in[2];  // [1024,1024]
    const float* Bm    = (const float*)d_in[3];  // [1024,24]
    const float* tau   = (const float*)d_in[4];  // [1024]
    const float* dsign = (const float*)d_in[5];  // [1024]
    float* out = (float*)d_out;                  // [64,500,1024] ++ [1,64,1024]
    float* ws  = (float*)d_ws;                   // 2 state buffers + barrier counter

    rnn_init_kernel<<<BATCH, 256, 0, stream>>>(r0, u, ws);
    rnn_persistent_kernel<<<NWG, 256, LDS_BYTES, stream>>>(u, W, Bm, tau, dsign, out, ws);
}